// GCN_18348100288800
// MI455X (gfx1250) — compile-verified
//
#include <hip/hip_runtime.h>

#define N_NODES 100000
#define N_EDGES 1600000
#define NFEAT 256
#define NHID 64
#define NOUT 16

typedef __attribute__((ext_vector_type(2))) float v2f;
typedef __attribute__((ext_vector_type(8))) float v8f;

// ---------------------------------------------------------------------------
// GEMM1: [N,256] x [256,64] -> [N,64] using V_WMMA_F32_16X16X4_F32 (full fp32)
// block = 256 threads = 8 waves; block tile 32(M) x 64(N); wave tile 16x16.
// K streamed through LDS in chunks of 64.
// ---------------------------------------------------------------------------
#define G1_KC 64
#define SA_STRIDE 72   // padded (floats): 288B row pitch -> 16B aligned, banks spread

__global__ __launch_bounds__(256) void gemm1_wmma(
    const float* __restrict__ A,   // embeddings [N_NODES, NFEAT]
    const float* __restrict__ B,   // W1 [NFEAT, NHID]
    float* __restrict__ C)         // [N_NODES, NHID]
{
  __shared__ float sA[32 * SA_STRIDE];   // 32 x 64 chunk of A (padded)
  __shared__ float sB[G1_KC * NHID];     // 64 x 64 chunk of W1

  const int tid  = threadIdx.x;
  const int wave = tid >> 5;
  const int lane = tid & 31;
  const int half = lane >> 4;   // 0: lanes 0-15, 1: lanes 16-31
  const int l    = lane & 15;

  const int tileM = wave >> 2;  // 0..1
  const int tileN = wave & 3;   // 0..3
  const int rowBase = blockIdx.x * 32;   // N_NODES % 32 == 0 -> no tail

  v8f acc = {};

  for (int kc = 0; kc < NFEAT; kc += G1_KC) {
    // cooperative load: A chunk 32x64 = 512 float4 (2 per thread), coalesced
    #pragma unroll
    for (int i = 0; i < 2; ++i) {
      int f  = tid + i * 256;
      int r  = f >> 4;
      int c4 = f & 15;
      float4 v = *(const float4*)(A + (size_t)(rowBase + r) * NFEAT + kc + c4 * 4);
      *(float4*)(sA + r * SA_STRIDE + c4 * 4) = v;
    }
    // cooperative load: W1 chunk 64x64 = 1024 float4 (4 per thread)
    #pragma unroll
    for (int i = 0; i < 4; ++i) {
      int f  = tid + i * 256;
      int k  = f >> 4;
      int c4 = f & 15;
      float4 v = *(const float4*)(B + (size_t)(kc + k) * NHID + c4 * 4);
      *(float4*)(sB + k * NHID + c4 * 4) = v;
    }
    __syncthreads();

    // 16 WMMA steps per chunk (K advances by 4 per v_wmma_f32_16x16x4_f32)
    #pragma unroll
    for (int k4 = 0; k4 < G1_KC; k4 += 4) {
      v2f a, b;
      const int r = tileM * 16 + l;
      // A 16x4 layout: VGPR0 = K{0|2}, VGPR1 = K{1|3} split by lane half
      a.x = sA[r * SA_STRIDE + k4 + 2 * half + 0];
      a.y = sA[r * SA_STRIDE + k4 + 2 * half + 1];
      // B 4x16 layout: rows striped across lanes within VGPR
      b.x = sB[(k4 + 2 * half + 0) * NHID + tileN * 16 + l];
      b.y = sB[(k4 + 2 * half + 1) * NHID + tileN * 16 + l];
      acc = __builtin_amdgcn_wmma_f32_16x16x4_f32(false, a, false, b,
                                                  (short)0, acc, false, false);
    }
    __syncthreads();
  }

  // C/D layout: VGPR i holds row (i + 8*half), col = lane&15
  #pragma unroll
  for (int i = 0; i < 8; ++i) {
    int row = rowBase + tileM * 16 + i + 8 * half;
    int col = tileN * 16 + l;
    C[(size_t)row * NHID + col] = acc[i];
  }
}

// ---------------------------------------------------------------------------
// GEMM2: [N,64] x [64,16] -> [N,16], one 16-row WMMA tile per wave,
// 128 rows per block, H staged through LDS, W2 resident in LDS.
// ---------------------------------------------------------------------------
#define G2_ROWS 128
#define SH_STRIDE 72

__global__ __launch_bounds__(256) void gemm2_wmma(
    const float* __restrict__ H,    // [N_NODES, NHID]
    const float* __restrict__ W2,   // [NHID, NOUT]
    float* __restrict__ C)          // [N_NODES, NOUT]
{
  __shared__ float sH[G2_ROWS * SH_STRIDE];  // 128 x 64 padded
  __shared__ float sW[NHID * NOUT];          // 64 x 16

  const int tid  = threadIdx.x;
  const int wave = tid >> 5;   // each wave owns one 16-row tile
  const int lane = tid & 31;
  const int half = lane >> 4;
  const int l    = lane & 15;
  const int rowBase = blockIdx.x * G2_ROWS;

  // W2: 1024 floats = 256 float4, one per thread
  {
    int k  = tid >> 2;
    int c4 = tid & 3;
    *(float4*)(sW + k * NOUT + c4 * 4) =
        *(const float4*)(W2 + k * NOUT + c4 * 4);
  }
  // H block: 128x64 floats = 2048 float4 (8 per thread); clamp tail rows
  #pragma unroll
  for (int i = 0; i < 8; ++i) {
    int f  = tid + i * 256;
    int r  = f >> 4;
    int c4 = f & 15;
    int grow = rowBase + r;
    if (grow >= N_NODES) grow = N_NODES - 1;   // clamped rows are never stored
    *(float4*)(sH + r * SH_STRIDE + c4 * 4) =
        *(const float4*)(H + (size_t)grow * NHID + c4 * 4);
  }
  __syncthreads();

  v8f acc = {};
  #pragma unroll
  for (int k4 = 0; k4 < NHID; k4 += 4) {
    v2f a, b;
    const int r = wave * 16 + l;
    a.x = sH[r * SH_STRIDE + k4 + 2 * half + 0];
    a.y = sH[r * SH_STRIDE + k4 + 2 * half + 1];
    b.x = sW[(k4 + 2 * half + 0) * NOUT + l];
    b.y = sW[(k4 + 2 * half + 1) * NOUT + l];
    acc = __builtin_amdgcn_wmma_f32_16x16x4_f32(false, a, false, b,
                                                (short)0, acc, false, false);
  }

  #pragma unroll
  for (int i = 0; i < 8; ++i) {
    int row = rowBase + wave * 16 + i + 8 * half;
    if (row < N_NODES) C[(size_t)row * NOUT + l] = acc[i];
  }
}

// ---------------------------------------------------------------------------
// SpMM over 64 features: one wave per edge, each lane owns a float2 of the row.
// Accumulator lives in L2 (25.6 MB << 192 MB) -> fast f32 atomics.
// ---------------------------------------------------------------------------
__global__ __launch_bounds__(256) void spmm_nhid_kernel(
    const float* __restrict__ X,    // [N_NODES, NHID]
    const int*   __restrict__ src,
    const int*   __restrict__ dst,
    const float* __restrict__ val,
    float*       __restrict__ Y)    // [N_NODES, NHID] pre-zeroed accumulator
{
  const int lane   = threadIdx.x & 31;
  const int waveId = (blockIdx.x * blockDim.x + threadIdx.x) >> 5;
  const int nWaves = (gridDim.x * blockDim.x) >> 5;
  for (int e = waveId; e < N_EDGES; e += nWaves) {
    const int   s = src[e];
    const int   d = dst[e];
    const float w = val[e];
    if (e + nWaves < N_EDGES)   // pipeline the next gather row
      __builtin_prefetch(X + (size_t)src[e + nWaves] * NHID + lane * 2, 0, 0);
    const float2 x = *(const float2*)(X + (size_t)s * NHID + lane * 2);
    unsafeAtomicAdd(Y + (size_t)d * NHID + lane * 2 + 0, w * x.x);
    unsafeAtomicAdd(Y + (size_t)d * NHID + lane * 2 + 1, w * x.y);
  }
}

// SpMM over 16 features: thread per (edge, feature)
__global__ __launch_bounds__(256) void spmm_nout_kernel(
    const float* __restrict__ X,    // [N_NODES, NOUT]
    const int*   __restrict__ src,
    const int*   __restrict__ dst,
    const float* __restrict__ val,
    float*       __restrict__ Y)    // [N_NODES, NOUT] pre-initialized to b2
{
  const long long total  = (long long)N_EDGES * NOUT;
  const long long stride = (long long)gridDim.x * blockDim.x;
  for (long long idx = (long long)blockIdx.x * blockDim.x + threadIdx.x;
       idx < total; idx += stride) {
    const int e = (int)(idx >> 4);
    const int f = (int)(idx & 15);
    unsafeAtomicAdd(Y + (size_t)dst[e] * NOUT + f,
                    val[e] * X[(size_t)src[e] * NOUT + f]);
  }
}

// ---------------------------------------------------------------------------
// Elementwise helpers
// ---------------------------------------------------------------------------
__global__ void zero_kernel(float* __restrict__ p, int n) {
  for (int i = blockIdx.x * blockDim.x + threadIdx.x; i < n;
       i += gridDim.x * blockDim.x)
    p[i] = 0.0f;
}

__global__ void bias_relu_dropout_kernel(float* __restrict__ h,
                                         const float* __restrict__ b1,
                                         const float* __restrict__ mask, int n) {
  for (int i = blockIdx.x * blockDim.x + threadIdx.x; i < n;
       i += gridDim.x * blockDim.x) {
    float v = h[i] + b1[i & (NHID - 1)];
    v = v > 0.0f ? v : 0.0f;
    h[i] = v * mask[i];
  }
}

__global__ void init_out_kernel(float* __restrict__ out,
                                const float* __restrict__ b2, int n) {
  for (int i = blockIdx.x * blockDim.x + threadIdx.x; i < n;
       i += gridDim.x * blockDim.x)
    out[i] = b2[i & (NOUT - 1)];
}

// ---------------------------------------------------------------------------
extern "C" void kernel_launch(void* const* d_in, const int* in_sizes, int n_in,
                              void* d_out, int out_size, void* d_ws, size_t ws_size,
                              hipStream_t stream) {
  const float* emb  = (const float*)d_in[0];
  const float* W1   = (const float*)d_in[1];
  const float* b1   = (const float*)d_in[2];
  const float* W2   = (const float*)d_in[3];
  const float* b2   = (const float*)d_in[4];
  const float* eval = (const float*)d_in[5];
  const float* mask = (const float*)d_in[6];
  const int*   esrc = (const int*)d_in[7];
  const int*   edst = (const int*)d_in[8];
  float* out = (float*)d_out;

  char* ws = (char*)d_ws;
  float* c1 = (float*)ws;                                    // [N,64] support1
  float* h  = (float*)(ws + (size_t)N_NODES * NHID * 4);     // [N,64] agg1 -> h
  float* c2 = c1;                                            // reuse: c1 dead after spmm1

  // 1. zero the first SpMM accumulator (fresh every call -> replay-safe)
  zero_kernel<<<2048, 256, 0, stream>>>(h, N_NODES * NHID);
  // 2. support1 = emb @ W1  (WMMA fp32)
  gemm1_wmma<<<N_NODES / 32, 256, 0, stream>>>(emb, W1, c1);
  // 3. agg1 = spmm(adj, support1)
  spmm_nhid_kernel<<<4096, 256, 0, stream>>>(c1, esrc, edst, eval, h);
  // 4. h = relu(agg1 + b1) * dropout_mask   (in place)
  bias_relu_dropout_kernel<<<2048, 256, 0, stream>>>(h, b1, mask, N_NODES * NHID);
  // 5. support2 = h @ W2  (WMMA fp32)
  gemm2_wmma<<<(N_NODES + G2_ROWS - 1) / G2_ROWS, 256, 0, stream>>>(h, W2, c2);
  // 6. out = b2 (broadcast)
  init_out_kernel<<<1024, 256, 0, stream>>>(out, b2, N_NODES * NOUT);
  // 7. out += spmm(adj, support2)
  spmm_nout_kernel<<<8192, 256, 0, stream>>>(c2, esrc, edst, eval, out);
}